// GINModel_59004260713104
// MI455X (gfx1250) — compile-verified
//
#include <hip/hip_runtime.h>
#include <hip/hip_bf16.h>

#define DF      128
#define NNODES  50000
#define NEDGES  640000
#define LAYERS  3
#define OUTF    2
#define BN_EPSF 1e-5f
#define PN_EPSF 1e-5f

// LDS row stride for the staged weight matrix: 144 dwords so rows k and k+2
// (the two EXEC-halves of a wave) land on disjoint 32-bank sets (288 % 64 == 32).
#define WSTRIDE 144

typedef float v2f __attribute__((ext_vector_type(2)));
typedef float v8f __attribute__((ext_vector_type(8)));
typedef int   gv4i __attribute__((vector_size(16)));   // matches builtin param type

#define AS1 __attribute__((address_space(1)))
#define AS3 __attribute__((address_space(3)))

#if defined(__has_builtin)
#if __has_builtin(__builtin_amdgcn_global_load_async_to_lds_b128) && \
    __has_builtin(__builtin_amdgcn_s_wait_asynccnt)
#define USE_ASYNC_LDS 1
#endif
#endif

// ---------------------------------------------------------------- utilities
__global__ void fill_zero_kernel(float* __restrict__ p, int n) {
  int i = blockIdx.x * blockDim.x + threadIdx.x;
  if (i < n) p[i] = 0.0f;
}

// ------------------------------------------------ z = (1+eps) * PN(h)  (elementwise)
__global__ void init_z_kernel(const float* __restrict__ h, float* __restrict__ z,
                              const float* __restrict__ pn_mu,
                              const float* __restrict__ pn_inv,
                              const float* __restrict__ eps_l, int apply_pn) {
  long long i = (long long)blockIdx.x * blockDim.x + threadIdx.x;  // float4 index
  const long long total = (long long)NNODES * DF / 4;
  if (i >= total) return;
  float e = 1.0f + eps_l[0];
  float4 v = ((const float4*)h)[i];
  if (apply_pn) {
    int d = (int)((i * 4) & (DF - 1));
    float inv = pn_inv[0];
    v.x = (v.x - pn_mu[d + 0]) * inv;
    v.y = (v.y - pn_mu[d + 1]) * inv;
    v.z = (v.z - pn_mu[d + 2]) * inv;
    v.w = (v.w - pn_mu[d + 3]) * inv;
  }
  v.x *= e; v.y *= e; v.z *= e; v.w *= e;
  ((float4*)z)[i] = v;
}

// ------------------------------------------------ z[dst] += PN(h[src])  (one wave/edge)
__global__ void scatter_kernel(const float* __restrict__ h,
                               const int* __restrict__ ei,
                               float* __restrict__ z,
                               const float* __restrict__ pn_mu,
                               const float* __restrict__ pn_inv, int apply_pn) {
  int wave = (blockIdx.x * blockDim.x + threadIdx.x) >> 5;
  int lane = threadIdx.x & 31;
  if (wave >= NEDGES) return;
  int src = ei[wave];
  int dst = ei[NEDGES + wave];
  int d = lane * 4;
  float4 v = *(const float4*)(h + (long long)src * DF + d);
  if (apply_pn) {
    float inv = pn_inv[0];
    v.x = (v.x - pn_mu[d + 0]) * inv;
    v.y = (v.y - pn_mu[d + 1]) * inv;
    v.z = (v.z - pn_mu[d + 2]) * inv;
    v.w = (v.w - pn_mu[d + 3]) * inv;
  }
  float* zp = z + (long long)dst * DF + d;
  atomicAdd(zp + 0, v.x);
  atomicAdd(zp + 1, v.y);
  atomicAdd(zp + 2, v.z);
  atomicAdd(zp + 3, v.w);
}

// ------------------------------------------------ WMMA GEMM:  out = op(A) @ W + bias
// 5 waves per block, one 16-row band per wave; W staged once per block into LDS
// (async global->LDS copy on CDNA5), B-fragments served from LDS.
// fuse_bn: A element k transformed as relu(scale[k]*a + shift[k]) on load.
// Epilogue accumulates per-column sum / sumsq for BatchNorm statistics.
__global__ void __launch_bounds__(160)
gemm_wmma_kernel(const float* __restrict__ A,
                 const float* __restrict__ W,
                 const float* __restrict__ bias,
                 float* __restrict__ out,
                 float* __restrict__ colsum,
                 float* __restrict__ colsumsq,
                 const float* __restrict__ scale,
                 const float* __restrict__ shift,
                 int fuse_bn) {
  __shared__ float wl[DF * WSTRIDE];          // 128 x 144 dwords = 73.7 KB

  const int tid = threadIdx.x;
  // ---- stage W[128][128] into LDS (row stride WSTRIDE), 16B chunks ----
  for (int idx = tid; idx < DF * (DF / 4); idx += 160) {
    const int r = idx >> 5;                   // row 0..127
    const int c = (idx & 31) * 4;             // col 0,4,...,124
    const float* gsrc = W + r * DF + c;
    float* ldst = &wl[r * WSTRIDE + c];
#if defined(USE_ASYNC_LDS)
    __builtin_amdgcn_global_load_async_to_lds_b128((AS1 gv4i*)gsrc,
                                                   (AS3 gv4i*)ldst, 0, 0);
#else
    *(float4*)ldst = *(const float4*)gsrc;
#endif
  }
#if defined(USE_ASYNC_LDS)
  __builtin_amdgcn_s_wait_asynccnt(0);
#endif
  __syncthreads();

  const int lane  = tid & 31;
  const int waveid = tid >> 5;                // 0..4
  const int rbase = (blockIdx.x * 5 + waveid) * 16;   // 625*5 = 3125 tiles exactly
  const int half  = lane >> 4;                // 0: lanes 0-15, 1: lanes 16-31
  const int lid   = lane & 15;

  v8f acc[8] = {};                            // 8 N-tiles of 16x16 f32 C/D

  for (int ks = 0; ks < 32; ++ks) {           // K = 128, 4 per WMMA
    const int k4 = ks * 4;
    const int ka = k4 + 2 * half;             // A/B frag K index for this half
    // A fragment: 16x4 f32 -> 2 VGPRs (rows stay hot in L0 across the ks loop)
    const float* arow = A + (long long)(rbase + lid) * DF + ka;
    v2f a;
    a.x = arow[0];
    a.y = arow[1];
    if (fuse_bn) {
      a.x = fmaxf(scale[ka + 0] * a.x + shift[ka + 0], 0.0f);
      a.y = fmaxf(scale[ka + 1] * a.y + shift[ka + 1], 0.0f);
    }
    const float* w0 = &wl[(ka + 0) * WSTRIDE];
    const float* w1 = &wl[(ka + 1) * WSTRIDE];
#pragma unroll
    for (int nt = 0; nt < 8; ++nt) {
      const int col = nt * 16 + lid;
      v2f b;                                  // 4x16 B frag rows ka, ka+1
      b.x = w0[col];
      b.y = w1[col];
      acc[nt] = __builtin_amdgcn_wmma_f32_16x16x4_f32(
          /*neg_a=*/false, a, /*neg_b=*/false, b,
          /*c_mod=*/(short)0, acc[nt], /*reuse_a=*/false, /*reuse_b=*/false);
    }
  }

  // Epilogue: bias, store, BN partial sums (lane pairs l / l+16 share a column)
#pragma unroll
  for (int nt = 0; nt < 8; ++nt) {
    const int col = nt * 16 + lid;
    const float bv = bias[col];
    float s = 0.0f, s2 = 0.0f;
#pragma unroll
    for (int r = 0; r < 8; ++r) {
      float y = acc[nt][r] + bv;
      const int row = rbase + r + half * 8;   // C layout: M = r (+8 for upper half)
      out[(long long)row * DF + col] = y;
      s  += y;
      s2 += y * y;
    }
    s  += __shfl_xor(s, 16);
    s2 += __shfl_xor(s2, 16);
    if (half == 0) {
      atomicAdd(&colsum[col], s);
      atomicAdd(&colsumsq[col], s2);
    }
  }
}

// ------------------------------------------------ BN stats -> affine (scale, shift)
__global__ void finalize_bn_kernel(const float* __restrict__ colsum,
                                   const float* __restrict__ colsumsq,
                                   const float* __restrict__ g,
                                   const float* __restrict__ b,
                                   float* __restrict__ scale,
                                   float* __restrict__ shift) {
  int c = threadIdx.x;
  float n = (float)NNODES;
  float mu  = colsum[c] / n;
  float var = colsumsq[c] / n - mu * mu;     // biased variance (torch BN)
  float s = g[c] * rsqrtf(var + BN_EPSF);
  scale[c] = s;
  shift[c] = b[c] - s * mu;
}

// ------------------------------------------------ h = relu(scale*z + shift); PN stats
__global__ void bn_relu_store_kernel(const float* __restrict__ z,
                                     const float* __restrict__ scale,
                                     const float* __restrict__ shift,
                                     float* __restrict__ h,
                                     float* __restrict__ colsum,
                                     float* __restrict__ totsumsq) {
  __shared__ float red[DF];
  const int c = threadIdx.x;                 // one column per thread
  const float s = scale[c], t = shift[c];
  const int r0 = blockIdx.x * 64;
  float ls = 0.0f, ls2 = 0.0f;
  for (int r = 0; r < 64; ++r) {
    int row = r0 + r;
    if (row >= NNODES) break;                // uniform across block
    float y = fmaxf(s * z[(long long)row * DF + c] + t, 0.0f);
    h[(long long)row * DF + c] = y;
    ls  += y;
    ls2 += y * y;
  }
  atomicAdd(&colsum[c], ls);
  red[c] = ls2;
  __syncthreads();
  for (int off = 64; off > 0; off >>= 1) {
    if (c < off) red[c] += red[c + off];
    __syncthreads();
  }
  if (c == 0) atomicAdd(totsumsq, red[0]);
}

// ------------------------------------------------ PairNorm mean/denominator
__global__ void finalize_pn_kernel(const float* __restrict__ colsum,
                                   const float* __restrict__ totsumsq,
                                   float* __restrict__ pn_mu,
                                   float* __restrict__ pn_inv) {
  __shared__ float red[DF];
  int c = threadIdx.x;
  float n = (float)NNODES;
  float mu = colsum[c] / n;
  pn_mu[c] = mu;
  red[c] = mu * mu;
  __syncthreads();
  for (int off = 64; off > 0; off >>= 1) {
    if (c < off) red[c] += red[c + off];
    __syncthreads();
  }
  if (c == 0) {
    // sum over (h - mu)^2 = totsumsq - N * sum(mu^2); mean over rows
    float centered = totsumsq[0] - n * red[0];
    pn_inv[0] = rsqrtf(PN_EPSF + centered / n);
  }
}

// ------------------------------------------------ classifier: out = h @ Wc + bc (one wave/row)
__global__ void classifier_kernel(const float* __restrict__ h,
                                  const float* __restrict__ Wc,
                                  const float* __restrict__ bc,
                                  float* __restrict__ out) {
  int wave = (blockIdx.x * blockDim.x + threadIdx.x) >> 5;
  int lane = threadIdx.x & 31;
  if (wave >= NNODES) return;
  const float* hr = h + (long long)wave * DF;
  int k = lane * 4;
  float4 v = *(const float4*)(hr + k);
  float o0 = v.x * Wc[(k + 0) * 2 + 0] + v.y * Wc[(k + 1) * 2 + 0] +
             v.z * Wc[(k + 2) * 2 + 0] + v.w * Wc[(k + 3) * 2 + 0];
  float o1 = v.x * Wc[(k + 0) * 2 + 1] + v.y * Wc[(k + 1) * 2 + 1] +
             v.z * Wc[(k + 2) * 2 + 1] + v.w * Wc[(k + 3) * 2 + 1];
  for (int off = 16; off > 0; off >>= 1) {
    o0 += __shfl_down(o0, off);
    o1 += __shfl_down(o1, off);
  }
  if (lane == 0) {
    out[(long long)wave * 2 + 0] = o0 + bc[0];
    out[(long long)wave * 2 + 1] = o1 + bc[1];
  }
}

// ---------------------------------------------------------------- host
extern "C" void kernel_launch(void* const* d_in, const int* in_sizes, int n_in,
                              void* d_out, int out_size, void* d_ws, size_t ws_size,
                              hipStream_t stream) {
  const float* x    = (const float*)d_in[0];
  const int*   ei   = (const int*)d_in[1];       // (2, E) int32
  const float* W1   = (const float*)d_in[2];     // (L, D, D)
  const float* b1   = (const float*)d_in[3];     // (L, D)
  const float* g1   = (const float*)d_in[4];
  const float* be1  = (const float*)d_in[5];
  const float* W2   = (const float*)d_in[6];
  const float* b2   = (const float*)d_in[7];
  const float* bng  = (const float*)d_in[8];
  const float* bnb  = (const float*)d_in[9];
  const float* eps  = (const float*)d_in[10];    // (L,)
  const float* clsW = (const float*)d_in[11];    // (D, 2)
  const float* clsb = (const float*)d_in[12];    // (2,)
  float* out = (float*)d_out;

  const long long ND = (long long)NNODES * DF;
  float* ws   = (float*)d_ws;
  float* zbuf = ws;                 // [N, D]
  float* ybuf = ws + ND;            // [N, D]
  float* hbuf = ws + 2 * ND;        // [N, D]
  float* stat = ws + 3 * ND;
  float* colsum   = stat;           // 128
  float* colsumsq = stat + 128;     // 128
  float* totsumsq = stat + 256;     // 1    (contiguous with the sums -> 1 clear)
  float* scale    = stat + 257;     // 128
  float* shift    = stat + 385;     // 128
  float* pn_mu    = stat + 513;     // 128
  float* pn_inv   = stat + 641;     // 1

  const dim3 blk256(256), blk128(128), blk160(160);
  const int initBlocks    = (int)((ND / 4 + 255) / 256);       // float4 elementwise
  const int scatterBlocks = (NEDGES * 32 + 255) / 256;
  const int gemmBlocks    = NNODES / 16 / 5;                    // 625 (5 waves/block)
  const int bnBlocks      = (NNODES + 63) / 64;                 // 782
  const int clsBlocks     = (NNODES * 32 + 255) / 256;          // 6250

  int apply_pn = 0;
  for (int l = 0; l < LAYERS; ++l) {
    const float* hcur = (l == 0) ? x : hbuf;

    // z = (1+eps_l) * PN(h) ; then z[dst] += PN(h[src])
    init_z_kernel<<<initBlocks, blk256, 0, stream>>>(hcur, zbuf, pn_mu, pn_inv,
                                                     eps + l, apply_pn);
    scatter_kernel<<<scatterBlocks, blk256, 0, stream>>>(hcur, ei, zbuf, pn_mu,
                                                         pn_inv, apply_pn);

    // GEMM1: y = z @ W1 + b1 (+ BN1 stats)
    fill_zero_kernel<<<2, blk256, 0, stream>>>(colsum, 257);
    gemm_wmma_kernel<<<gemmBlocks, blk160, 0, stream>>>(
        zbuf, W1 + (long long)l * DF * DF, b1 + l * DF, ybuf,
        colsum, colsumsq, scale, shift, /*fuse_bn=*/0);
    finalize_bn_kernel<<<1, blk128, 0, stream>>>(colsum, colsumsq,
                                                 g1 + l * DF, be1 + l * DF,
                                                 scale, shift);

    // GEMM2 (fused BN1+ReLU on A load): z = relu(bn1(y)) @ W2 + b2 (+ BN2 stats)
    fill_zero_kernel<<<2, blk256, 0, stream>>>(colsum, 257);
    gemm_wmma_kernel<<<gemmBlocks, blk160, 0, stream>>>(
        ybuf, W2 + (long long)l * DF * DF, b2 + l * DF, zbuf,
        colsum, colsumsq, scale, shift, /*fuse_bn=*/1);
    finalize_bn_kernel<<<1, blk128, 0, stream>>>(colsum, colsumsq,
                                                 bng + l * DF, bnb + l * DF,
                                                 scale, shift);

    // h = relu(bn2(z)) ; accumulate PairNorm stats
    fill_zero_kernel<<<2, blk256, 0, stream>>>(colsum, 257);
    bn_relu_store_kernel<<<bnBlocks, blk128, 0, stream>>>(zbuf, scale, shift,
                                                          hbuf, colsum, totsumsq);

    if (l < LAYERS - 1) {
      finalize_pn_kernel<<<1, blk128, 0, stream>>>(colsum, totsumsq, pn_mu, pn_inv);
      apply_pn = 1;                  // PN applied lazily at next layer's loads
    }
  }

  classifier_kernel<<<clsBlocks, blk256, 0, stream>>>(hbuf, clsW, clsb, out);
}